// local_branch_70111046140245
// MI455X (gfx1250) — compile-verified
//
#include <hip/hip_runtime.h>

// MI455X / gfx1250, wave32. All matrix work uses V_WMMA_F32_16X16X4_F32
// (native fp32 WMMA) chained over K. Problem is ~0.28 GFLOP / ~3MB: launch-
// latency bound, fits in L2, so we fuse algebraically (out_t is never
// materialized) and keep full fp32 precision.

typedef __attribute__((ext_vector_type(2))) float v2f;
typedef __attribute__((ext_vector_type(8))) float v8f;

#define B_   2
#define CIN  256
#define DK_  128
#define NP   1024   // HK*WK
#define NI   256    // HQ*WQ
#define OOUT 12

__device__ __forceinline__ v8f wmma_f32_k4(v2f a, v2f b, v8f c) {
  // 8 args: (neg_a, A, neg_b, B, c_mod, C, reuse_a, reuse_b)
  return __builtin_amdgcn_wmma_f32_16x16x4_f32(false, a, false, b, (short)0, c,
                                               false, false);
}

// ---------------------------------------------------------------------------
// Kernel 1: key_f[b] = W_key(128x256) @ k[b](256x1024)
// One wave per 16x16 output tile; K loop in steps of 4.
// Fragment layout (ISA 7.12.2):
//   A 16x4 f32 : lane m=l%16, half=l/16 -> a = {A[m][2*half], A[m][2*half+1]}
//   B 4x16 f32 : lane n=l%16, half     -> b = {B[2*half][n],  B[2*half+1][n]}
//   C/D 16x16  : lane n=l%16; VGPR v holds M = v + 8*half
// ---------------------------------------------------------------------------
__global__ __launch_bounds__(256) void keyf_kernel(const float* __restrict__ k,
                                                   const float* __restrict__ Wkey,
                                                   float* __restrict__ keyf) {
  const int lane = threadIdx.x & 31;
  const int wave = threadIdx.x >> 5;
  const int wid  = blockIdx.x * 8 + wave;   // 0..1023
  const int b    = wid >> 9;                // batch
  const int rem  = wid & 511;
  const int mt   = rem >> 6;                // 0..7   (o tiles, 128 rows)
  const int nt   = rem & 63;                // 0..63  (p tiles, 1024 cols)
  const int half = lane >> 4, lid = lane & 15;
  const int row0 = mt * 16, col0 = nt * 16;

  const float* A  = Wkey;                   // 128 x 256
  const float* Bm = k + (size_t)b * CIN * NP; // 256 x 1024

  v8f acc = {};
#pragma unroll 8
  for (int kk = 0; kk < CIN; kk += 4) {
    v2f a, bb;
    const float* ap = A + (row0 + lid) * CIN + kk + 2 * half;
    a.x = ap[0];
    a.y = ap[1];
    const float* bp = Bm + (size_t)(kk + 2 * half) * NP + col0 + lid;
    bb.x = bp[0];
    bb.y = bp[NP];
    acc = wmma_f32_k4(a, bb, acc);
  }
  float* D = keyf + (size_t)b * DK_ * NP;
#pragma unroll
  for (int v = 0; v < 8; ++v)
    D[(size_t)(row0 + v + 8 * half) * NP + col0 + lid] = acc[v];
}

// ---------------------------------------------------------------------------
// Kernel 2: s[b,p] = sum_c w3[c]*key_f[b,c,p];  c[b,o] = sum_p Wobj[o,p]*(s+b3)
// Tiny; one block per batch, s staged through LDS for the c reduction.
// ---------------------------------------------------------------------------
__global__ __launch_bounds__(256) void prep_kernel(const float* __restrict__ keyf,
                                                   const float* __restrict__ w3,
                                                   const float* __restrict__ b3,
                                                   const float* __restrict__ Wobj,
                                                   float* __restrict__ s,
                                                   float* __restrict__ cvec) {
  const int b = blockIdx.x;
  __shared__ float s_sh[NP];
  const float* kf = keyf + (size_t)b * DK_ * NP;
  for (int p = threadIdx.x; p < NP; p += 256) {
    float acc = 0.0f;
#pragma unroll 8
    for (int cc = 0; cc < DK_; ++cc)
      acc = fmaf(w3[cc], kf[(size_t)cc * NP + p], acc);
    s_sh[p] = acc;
    s[b * NP + p] = acc;
  }
  __syncthreads();
  if (threadIdx.x < 16) {
    const int o = threadIdx.x;
    float acc = 0.0f;
    if (o < OOUT) {
      const float bb = b3[0];
      for (int p = 0; p < NP; ++p)
        acc = fmaf(Wobj[o * NP + p], s_sh[p] + bb, acc);
    }
    cvec[b * 16 + o] = acc;  // rows 12..15 stay 0 (padding)
  }
}

// ---------------------------------------------------------------------------
// Kernel 3: att[b] = q[b]^T(256x128) @ key_f[b](128x1024)
// A is read transposed from q (q stored as (d,i)).
// ---------------------------------------------------------------------------
__global__ __launch_bounds__(256) void att_kernel(const float* __restrict__ q,
                                                  const float* __restrict__ keyf,
                                                  float* __restrict__ att) {
  const int lane = threadIdx.x & 31;
  const int wave = threadIdx.x >> 5;
  const int wid  = blockIdx.x * 8 + wave;   // 0..2047
  const int b    = wid >> 10;
  const int rem  = wid & 1023;
  const int mt   = rem >> 6;                // 0..15 (i tiles, 256 rows)
  const int nt   = rem & 63;                // 0..63 (p tiles)
  const int half = lane >> 4, lid = lane & 15;
  const int row0 = mt * 16, col0 = nt * 16;

  const float* Q = q    + (size_t)b * DK_ * NI;  // (d, i) = 128 x 256
  const float* K = keyf + (size_t)b * DK_ * NP;  // (d, p) = 128 x 1024

  v8f acc = {};
#pragma unroll 8
  for (int kk = 0; kk < DK_; kk += 4) {
    v2f a, bb;
    // A[i,d] = Q[d,i]
    const float* ap = Q + (size_t)(kk + 2 * half) * NI + row0 + lid;
    a.x = ap[0];
    a.y = ap[NI];
    const float* bp = K + (size_t)(kk + 2 * half) * NP + col0 + lid;
    bb.x = bp[0];
    bb.y = bp[NP];
    acc = wmma_f32_k4(a, bb, acc);
  }
  float* D = att + (size_t)b * NI * NP;
#pragma unroll
  for (int v = 0; v < 8; ++v)
    D[(size_t)(row0 + v + 8 * half) * NP + col0 + lid] = acc[v];
}

// ---------------------------------------------------------------------------
// Kernel 4: final[b,o,i] = sum_p (Wobj[o,p]*s[b,p]) * att[b,i,p] + c[b,o]
// M=12 padded to 16 (A rows >=12 are zero via select, no EXEC divergence
// around the WMMA). B fragment reads att transposed (B[p,i] = att[i,p]).
// ---------------------------------------------------------------------------
__global__ __launch_bounds__(256) void obj_kernel(const float* __restrict__ att,
                                                  const float* __restrict__ Wobj,
                                                  const float* __restrict__ s,
                                                  const float* __restrict__ cvec,
                                                  float* __restrict__ out) {
  const int lane = threadIdx.x & 31;
  const int wave = threadIdx.x >> 5;
  const int wid  = blockIdx.x * 8 + wave;   // 0..31
  const int b    = wid >> 4;
  const int nt   = wid & 15;                // i tile
  const int half = lane >> 4, lid = lane & 15;
  const int col0 = nt * 16;

  const float* ATT = att + (size_t)b * NI * NP;
  const float* S   = s + b * NP;
  const int o = (lid < OOUT) ? lid : (OOUT - 1);  // clamp to avoid OOB loads
  const bool valid = (lid < OOUT);

  v8f acc = {};
#pragma unroll 4
  for (int kk = 0; kk < NP; kk += 4) {
    const int p = kk + 2 * half;
    v2f a, bb;
    const float w0 = Wobj[o * NP + p];
    const float w1 = Wobj[o * NP + p + 1];
    a.x = valid ? w0 * S[p] : 0.0f;
    a.y = valid ? w1 * S[p + 1] : 0.0f;
    // B[p,i] = att[i,p]
    const float* bp = ATT + (size_t)(col0 + lid) * NP + p;
    bb.x = bp[0];
    bb.y = bp[1];
    acc = wmma_f32_k4(a, bb, acc);
  }
#pragma unroll
  for (int v = 0; v < 8; ++v) {
    const int row = v + 8 * half;
    if (row < OOUT)
      out[(size_t)b * OOUT * NI + row * NI + col0 + lid] =
          acc[v] + cvec[b * 16 + row];
  }
}

// ---------------------------------------------------------------------------
extern "C" void kernel_launch(void* const* d_in, const int* in_sizes, int n_in,
                              void* d_out, int out_size, void* d_ws, size_t ws_size,
                              hipStream_t stream) {
  const float* k    = (const float*)d_in[0];
  const float* q    = (const float*)d_in[1];
  const float* Wkey = (const float*)d_in[2];
  const float* w3   = (const float*)d_in[3];
  const float* b3   = (const float*)d_in[4];
  const float* Wobj = (const float*)d_in[5];

  float* ws   = (float*)d_ws;
  float* keyf = ws;                    // B*128*1024 = 262144 floats
  float* att  = keyf + B_ * DK_ * NP;  // B*256*1024 = 524288 floats
  float* s    = att + B_ * NI * NP;    // B*1024
  float* cvec = s + B_ * NP;           // B*16
  // total workspace: ~3.15 MB

  keyf_kernel<<<128, 256, 0, stream>>>(k, Wkey, keyf);
  prep_kernel<<<B_, 256, 0, stream>>>(keyf, w3, b3, Wobj, s, cvec);
  att_kernel<<<256, 256, 0, stream>>>(q, keyf, att);
  obj_kernel<<<4, 256, 0, stream>>>(att, Wobj, s, cvec, (float*)d_out);
}